// FrameAlignment_11888469475489
// MI455X (gfx1250) — compile-verified
//
#include <hip/hip_runtime.h>

typedef float v2f __attribute__((ext_vector_type(2)));
typedef float v4f __attribute__((ext_vector_type(4)));
typedef float v8f __attribute__((ext_vector_type(8)));
typedef unsigned int u32x4 __attribute__((ext_vector_type(4)));
typedef int i32x4 __attribute__((ext_vector_type(4)));
typedef int i32x8 __attribute__((ext_vector_type(8)));

#define B_   16
#define S_   2048
#define F_   64
#define JC   32          // j-chunk per WMMA sweep (8 x K=4 steps)
#define WST  36          // LDS row stride: 16B aligned, conflict-free for 16 rows

__global__ __launch_bounds__(128) void FrameAlignment_kernel(
    const float* __restrict__ score,    // [B,S,1]
    const float* __restrict__ feature,  // [B,S,F]
    const float* __restrict__ tmpl,     // [1,S,1]
    float* __restrict__ aligned,        // [B,S,F]   (d_out)
    float* __restrict__ path)           // [B,S,S]   (d_out + B*S*F)
{
    __shared__ float t_sh[S_];            // template, 8 KB (filled by TDM)
    __shared__ float wbuf[16 * WST];      // normalized weight tile (A matrix staging)
    __shared__ float red[128];            // reduction scratch
    __shared__ float s_sh[16], m_sh[16], rz_sh[16];

    const int tid = threadIdx.x;
    const int blk = blockIdx.x;
    const int b   = blk >> 7;             // / (S/16) == /128
    const int i0  = (blk & 127) << 4;     // first of 16 rows this block owns

    // ---- template -> LDS via Tensor Data Mover (ISA 08 §8.3/8.4) ----
    if (tid < 32) {                       // wave 0 issues the DMA (wave-uniform branch)
        const unsigned long long ga = (unsigned long long)(uintptr_t)tmpl;
        const unsigned int lds_off  = (unsigned int)(uintptr_t)t_sh; // flat LDS addr: low 32b = offset

        u32x4 g0;
        g0.x = 1u;                                    // count=1, user mode, no gather
        g0.y = lds_off;                               // D#.lds_addr (bytes)
        g0.z = (unsigned int)(ga & 0xFFFFFFFFu);      // global_addr[31:0]
        g0.w = (unsigned int)((ga >> 32) & 0x1FFFFFFu)// global_addr[56:32]
             | (2u << 30);                            // type = 2 ("image")

        i32x8 g1;
        g1[0] = (2 << 16);                            // wg_mask=0 (not in cluster), data_size=4B
        g1[1] = (int)(2048u << 16);                   // tensor_dim0 = 2048 (low 16 -> bits 63:48)
        g1[2] = (int)(1u << 16);                      // tensor_dim0 hi=0 ; tensor_dim1 = 1
        g1[3] = (int)(2048u << 16);                   // tensor_dim1 hi=0 ; tile_dim0 = 2048
        g1[4] = 0;                                    // tile_dim1/2 unused
        g1[5] = 2048;                                 // tensor_dim0_stride = 2048 elements
        g1[6] = 0;
        g1[7] = 0;

        i32x4 gz4 = {};                               // groups 2/3 unused (<=2D tensor)
        i32x8 gz8 = {};
        __builtin_amdgcn_tensor_load_to_lds(g0, g1, gz4, gz4, gz8, 0);
        __builtin_amdgcn_s_wait_tensorcnt(0);         // per-wave TENSORcnt: drain before barrier
    } else if (tid < 48) {
        s_sh[tid - 32] = score[(size_t)b * S_ + i0 + (tid - 32)];
    }
    __syncthreads();

    // ---- phase 1: m_r = min_j |s_r - t_j| (8 threads per row) ----
    {
        const int r = tid & 15, g = tid >> 4;
        const float s = s_sh[r];
        float mn = 3.4e38f;
        for (int j = g; j < S_; j += 8) mn = fminf(mn, fabsf(s - t_sh[j]));
        red[tid] = mn;
        __syncthreads();
        if (tid < 16) {
            float m = red[tid];
            #pragma unroll
            for (int g2 = 1; g2 < 8; ++g2) m = fminf(m, red[g2 * 16 + tid]);
            m_sh[tid] = m;
        }
        __syncthreads();
    }

    // ---- phase 2: Z_r = sum_j exp(m_r - |s_r - t_j|) ----
    {
        const int r = tid & 15, g = tid >> 4;
        const float s = s_sh[r], m = m_sh[r];
        float z = 0.f;
        for (int j = g; j < S_; j += 8) z += __expf(m - fabsf(s - t_sh[j]));
        red[tid] = z;
        __syncthreads();
        if (tid < 16) {
            float zz = red[tid];
            #pragma unroll
            for (int g2 = 1; g2 < 8; ++g2) zz += red[g2 * 16 + tid];
            rz_sh[tid] = 1.0f / zz;
        }
        __syncthreads();
    }

    // ---- phase 3: stream j; write path (NT); accumulate aligned via fp32 WMMA ----
    const int lane = tid & 31;
    const int wave = tid >> 5;
    const int n0   = wave * 16;           // each wave owns 16 of F=64 columns
    const int mrow = lane & 15;           // A-matrix row / B-matrix column (N)
    const int koff = (lane >> 4) * 2;     // hi-half lanes hold K+2,K+3

    v8f acc = {};                         // 16x16 fp32 C/D tile (8 VGPRs)
    const float* featB  = feature + (size_t)b * S_ * F_;
    float*       pathRw = path + ((size_t)b * S_ + i0) * (size_t)S_;

    const int pr = tid >> 3;              // 0..15 : row for weight compute
    const int pc = (tid & 7) * 4;         // 0..28 : 4 j's per thread

    for (int j0 = 0; j0 < S_; j0 += JC) {
        // compute normalized softmax weights, stage in LDS, stream to path
        {
            const float s = s_sh[pr], m = m_sh[pr], rz = rz_sh[pr];
            v4f p;
            p.x = __expf(m - fabsf(s - t_sh[j0 + pc + 0])) * rz;
            p.y = __expf(m - fabsf(s - t_sh[j0 + pc + 1])) * rz;
            p.z = __expf(m - fabsf(s - t_sh[j0 + pc + 2])) * rz;
            p.w = __expf(m - fabsf(s - t_sh[j0 + pc + 3])) * rz;
            *(v4f*)&wbuf[pr * WST + pc] = p;
            // 268 MB pure-output stream > 192 MB L2: bypass with non-temporal hint
            __builtin_nontemporal_store(p, (v4f*)&pathRw[(size_t)pr * S_ + j0 + pc]);
        }
        __syncthreads();

        // prefetch next chunk's feature rows (gfx1250 global_prefetch path)
        if (j0 + JC < S_)
            __builtin_prefetch(featB + (size_t)(j0 + JC + koff) * F_ + n0 + mrow, 0, 1);

        // 8 x V_WMMA_F32_16X16X4_F32 over this K-chunk
        #pragma unroll
        for (int k = 0; k < JC; k += 4) {
            v2f a, bb;
            a.x = wbuf[mrow * WST + k + koff + 0];
            a.y = wbuf[mrow * WST + k + koff + 1];
            const float* fp = featB + (size_t)(j0 + k + koff) * F_ + n0 + mrow;
            bb.x = fp[0];        // K = k+koff   , N = n0+mrow
            bb.y = fp[F_];       // K = k+koff+1 , N = n0+mrow
            acc = __builtin_amdgcn_wmma_f32_16x16x4_f32(
                      false, a, false, bb, (short)0, acc, false, false);
        }
        __syncthreads();
    }

    // ---- store aligned 16x16 tile per wave ----
    #pragma unroll
    for (int v = 0; v < 8; ++v) {
        const int row = i0 + v + (lane >> 4) * 8;        // C/D layout: M = v (+8 for hi lanes)
        aligned[((size_t)b * S_ + row) * F_ + n0 + (lane & 15)] = acc[v];
    }
}

extern "C" void kernel_launch(void* const* d_in, const int* in_sizes, int n_in,
                              void* d_out, int out_size, void* d_ws, size_t ws_size,
                              hipStream_t stream) {
    const float* score   = (const float*)d_in[0];   // [B,S,1]
    const float* feature = (const float*)d_in[1];   // [B,S,F]
    const float* tmpl    = (const float*)d_in[2];   // [1,S,1]

    float* aligned = (float*)d_out;                         // B*S*F
    float* path    = (float*)d_out + (size_t)B_ * S_ * F_;  // B*S*S

    dim3 grid(B_ * (S_ / 16));   // 2048 blocks
    dim3 block(128);             // 4 waves (wave32)
    FrameAlignment_kernel<<<grid, block, 0, stream>>>(score, feature, tmpl, aligned, path);
}